// GCN_7378753814586
// MI455X (gfx1250) — compile-verified
//
#include <hip/hip_runtime.h>

typedef __attribute__((ext_vector_type(2))) float v2f;
typedef __attribute__((ext_vector_type(8))) float v8f;

// ---------------------------------------------------------------------------
// Degree / normalization kernels
// ---------------------------------------------------------------------------
__global__ __launch_bounds__(256) void deg_init(float* __restrict__ deg, int n) {
    int i = blockIdx.x * blockDim.x + threadIdx.x;
    if (i < n) deg[i] = 1.0f;  // self-loop contributes 1
}

__global__ __launch_bounds__(256) void deg_count(const int* __restrict__ dst,
                                                 float* __restrict__ deg, int e) {
    int i = blockIdx.x * blockDim.x + threadIdx.x;
    if (i < e) unsafeAtomicAdd(&deg[dst[i]], 1.0f);
}

__global__ __launch_bounds__(256) void deg_finalize(float* __restrict__ dinv, int n) {
    int i = blockIdx.x * blockDim.x + threadIdx.x;
    if (i < n) dinv[i] = rsqrtf(dinv[i]);  // deg >= 1 always (self-loops)
}

__global__ __launch_bounds__(256) void zero_f32(float* __restrict__ p, size_t n) {
    size_t i = (size_t)blockIdx.x * blockDim.x + threadIdx.x;
    if (i < n) p[i] = 0.0f;
}

// ---------------------------------------------------------------------------
// fp32 WMMA GEMM: C[M,N] = A[M,K] @ B[K,N]  (+ optional bias & ReLU)
// One wave per 16x16 output tile; K-loop of V_WMMA_F32_16X16X4_F32.
// A layout (16x4 f32): lanes 0-15 -> K={0,1}, lanes 16-31 -> K={2,3}; M = lane&15
// B layout (4x16 f32): VGPR0 rows K=0/K=2, VGPR1 rows K=1/K=3; N = lane&15
// C/D layout: element (M = v + 8*(lane>>4), N = lane&15) in acc[v]
// ---------------------------------------------------------------------------
__global__ __launch_bounds__(256) void wmma_gemm_f32(
    const float* __restrict__ A, const float* __restrict__ B,
    const float* __restrict__ bias, float* __restrict__ C,
    int M, int N, int K, int colTiles, int numTiles, int doReluBias)
{
    int wave = threadIdx.x >> 5;
    int lane = threadIdx.x & 31;
    int tile = blockIdx.x * 8 + wave;
    if (tile >= numTiles) return;  // whole wave exits together: EXEC all-1 for WMMA

    int rt = tile / colTiles;
    int ct = tile - rt * colTiles;
    int l15   = lane & 15;
    int khalf = (lane >> 4) << 1;       // 0 or 2
    int row   = rt * 16 + l15;
    int rowc  = row < M ? row : M - 1;  // clamp A loads (results masked on store)
    int col   = ct * 16 + l15;
    int colc  = col < N ? col : N - 1;  // clamp B loads

    const float* Ap = A + (size_t)rowc * K + khalf;
    const float* Bp = B + (size_t)khalf * N + colc;

    v8f acc = {};
    for (int k0 = 0; k0 < K; k0 += 4) {
        v2f a; a.x = Ap[0]; a.y = Ap[1];
        v2f b; b.x = Bp[0]; b.y = Bp[N];
        acc = __builtin_amdgcn_wmma_f32_16x16x4_f32(
            /*neg_a=*/false, a, /*neg_b=*/false, b,
            /*c_mod=*/(short)0, acc, /*reuse_a=*/false, /*reuse_b=*/false);
        Ap += 4;
        Bp += (size_t)4 * N;
    }

    if (col < N) {
        int mbase = rt * 16 + (lane >> 4) * 8;
        float bb = doReluBias ? bias[col] : 0.0f;
#pragma unroll
        for (int v = 0; v < 8; ++v) {
            int mrow = mbase + v;
            if (mrow < M) {
                float val = acc[v] + bb;
                if (doReluBias) val = fmaxf(val, 0.0f);
                C[(size_t)mrow * N + col] = val;
            }
        }
    }
}

// ---------------------------------------------------------------------------
// Edge scatter: agg[dst] += xw[src] * dinv[src]*dinv[dst]
// One wave per edge. 128-feature variant: float4 gather, 4 fp32 atomics/lane.
// ---------------------------------------------------------------------------
__global__ __launch_bounds__(256) void scatter_feat128(
    const float* __restrict__ xw, const int* __restrict__ src,
    const int* __restrict__ dst, const float* __restrict__ dinv,
    float* __restrict__ agg, int E)
{
    int e = blockIdx.x * 8 + (threadIdx.x >> 5);
    if (e >= E) return;
    int lane = threadIdx.x & 31;
    int s = src[e], d = dst[e];
    float w = dinv[s] * dinv[d];
    const float4* xp = (const float4*)(xw + (size_t)s * 128);
    float4 v = xp[lane];
    float* ap = agg + (size_t)d * 128 + lane * 4;
    unsafeAtomicAdd(ap + 0, v.x * w);
    unsafeAtomicAdd(ap + 1, v.y * w);
    unsafeAtomicAdd(ap + 2, v.z * w);
    unsafeAtomicAdd(ap + 3, v.w * w);
}

// Narrow variant (outC in [33,64], here 40): lanes cover f=lane and f=lane+32.
__global__ __launch_bounds__(256) void scatter_feat_narrow(
    const float* __restrict__ xw, const int* __restrict__ src,
    const int* __restrict__ dst, const float* __restrict__ dinv,
    float* __restrict__ agg, int E, int outC)
{
    int e = blockIdx.x * 8 + (threadIdx.x >> 5);
    if (e >= E) return;
    int lane = threadIdx.x & 31;
    int s = src[e], d = dst[e];
    float w = dinv[s] * dinv[d];
    const float* xp = xw + (size_t)s * outC;
    float* ap = agg + (size_t)d * outC;
    unsafeAtomicAdd(ap + lane, xp[lane] * w);
    int f1 = lane + 32;
    if (f1 < outC) unsafeAtomicAdd(ap + f1, xp[f1] * w);
}

// ---------------------------------------------------------------------------
// h2 = relu(agg + xw * dinv^2 (self-loop) + bias)   [128 features]
// ---------------------------------------------------------------------------
__global__ __launch_bounds__(256) void finalize_relu128(
    const float* __restrict__ agg, const float* __restrict__ xw,
    const float* __restrict__ dinv, const float* __restrict__ bias,
    float* __restrict__ out, int nNodes)
{
    size_t idx = (size_t)blockIdx.x * blockDim.x + threadIdx.x;
    size_t total = (size_t)nNodes * 128;
    if (idx >= total) return;
    int i = (int)(idx >> 7);
    int f = (int)(idx & 127);
    float di = dinv[i];
    out[idx] = fmaxf(agg[idx] + xw[idx] * di * di + bias[f], 0.0f);
}

// ---------------------------------------------------------------------------
// row = agg + xw*dinv^2 + bias; out = log_softmax(row)  (wave per node, C<=64)
// ---------------------------------------------------------------------------
__global__ __launch_bounds__(256) void finalize_logsoftmax(
    const float* __restrict__ agg, const float* __restrict__ xw,
    const float* __restrict__ dinv, const float* __restrict__ bias,
    float* __restrict__ out, int nNodes, int C)
{
    int i = blockIdx.x * 8 + (threadIdx.x >> 5);
    if (i >= nNodes) return;
    int lane = threadIdx.x & 31;
    float di = dinv[i];
    float di2 = di * di;
    size_t base = (size_t)i * C;

    float v0 = agg[base + lane] + xw[base + lane] * di2 + bias[lane];  // C >= 32
    int f1 = lane + 32;
    bool has1 = f1 < C;
    float v1 = has1 ? (agg[base + f1] + xw[base + f1] * di2 + bias[f1])
                    : -__builtin_inff();

    float m = fmaxf(v0, v1);
#pragma unroll
    for (int off = 16; off >= 1; off >>= 1) m = fmaxf(m, __shfl_xor(m, off, 32));
    float s = expf(v0 - m) + (has1 ? expf(v1 - m) : 0.0f);
#pragma unroll
    for (int off = 16; off >= 1; off >>= 1) s += __shfl_xor(s, off, 32);
    float lse = m + logf(s);

    out[base + lane] = v0 - lse;
    if (has1) out[base + f1] = v1 - lse;
}

// ---------------------------------------------------------------------------
// Orchestration
// ---------------------------------------------------------------------------
extern "C" void kernel_launch(void* const* d_in, const int* in_sizes, int n_in,
                              void* d_out, int out_size, void* d_ws, size_t ws_size,
                              hipStream_t stream) {
    const float* x    = (const float*)d_in[0];
    const int*   ei   = (const int*)  d_in[1];
    const float* W1   = (const float*)d_in[2];
    const float* b1   = (const float*)d_in[3];
    const float* Wc1  = (const float*)d_in[4];
    const float* bc1  = (const float*)d_in[5];
    const float* Wc2  = (const float*)d_in[6];
    const float* bc2  = (const float*)d_in[7];

    const int HID    = 128;
    const int nNodes = in_sizes[0] / HID;
    const int E      = in_sizes[1] / 2;
    const int OUTC   = in_sizes[7];            // 40
    const int*   srcp = ei;
    const int*   dstp = ei + E;
    float* outp = (float*)d_out;

    // Workspace layout (floats). Buffers reused across phases.
    float* ws   = (float*)d_ws;
    float* dinv = ws;                                  // nNodes
    float* h1   = dinv + (size_t)nNodes;               // nNodes*128 (h1, later h2)
    float* xw   = h1   + (size_t)nNodes * HID;         // nNodes*128 (xw1, later xw2)
    float* agg  = xw   + (size_t)nNodes * HID;         // nNodes*128 (agg1, later agg2)

    const int TPB = 256;
    int mTiles = (nNodes + 15) / 16;

    // 1) dinv = rsqrt(deg + self-loop)
    deg_init    <<<(nNodes + TPB - 1) / TPB, TPB, 0, stream>>>(dinv, nNodes);
    deg_count   <<<(E + TPB - 1) / TPB, TPB, 0, stream>>>(dstp, dinv, E);
    deg_finalize<<<(nNodes + TPB - 1) / TPB, TPB, 0, stream>>>(dinv, nNodes);

    // 2) h1 = relu(x @ W1 + b1)  [nNodes,128]
    {
        int colT = HID / 16, tiles = mTiles * colT;
        wmma_gemm_f32<<<(tiles + 7) / 8, TPB, 0, stream>>>(
            x, W1, b1, h1, nNodes, HID, HID, colT, tiles, 1);
    }

    // 3) xw1 = h1 @ Wc1  [nNodes,128]
    {
        int colT = HID / 16, tiles = mTiles * colT;
        wmma_gemm_f32<<<(tiles + 7) / 8, TPB, 0, stream>>>(
            h1, Wc1, nullptr, xw, nNodes, HID, HID, colT, tiles, 0);
    }

    // 4) agg1 = scatter-add of normalized messages
    {
        size_t n = (size_t)nNodes * HID;
        zero_f32<<<(unsigned)((n + TPB - 1) / TPB), TPB, 0, stream>>>(agg, n);
        scatter_feat128<<<(E + 7) / 8, TPB, 0, stream>>>(xw, srcp, dstp, dinv, agg, E);
    }

    // 5) h2 = relu(agg1 + xw1*dinv^2 + bc1)  -> overwrite h1
    {
        size_t n = (size_t)nNodes * HID;
        finalize_relu128<<<(unsigned)((n + TPB - 1) / TPB), TPB, 0, stream>>>(
            agg, xw, dinv, bc1, h1, nNodes);
    }

    // 6) xw2 = h2 @ Wc2  [nNodes,40] -> reuse xw buffer
    {
        int colT = (OUTC + 15) / 16, tiles = mTiles * colT;
        wmma_gemm_f32<<<(tiles + 7) / 8, TPB, 0, stream>>>(
            h1, Wc2, nullptr, xw, nNodes, OUTC, HID, colT, tiles, 0);
    }

    // 7) agg2 scatter  -> reuse agg buffer
    {
        size_t n = (size_t)nNodes * OUTC;
        zero_f32<<<(unsigned)((n + TPB - 1) / TPB), TPB, 0, stream>>>(agg, n);
        scatter_feat_narrow<<<(E + 7) / 8, TPB, 0, stream>>>(
            xw, srcp, dstp, dinv, agg, E, OUTC);
    }

    // 8) out = log_softmax(agg2 + xw2*dinv^2 + bc2)
    finalize_logsoftmax<<<(nNodes + 7) / 8, TPB, 0, stream>>>(
        agg, xw, dinv, bc2, outp, nNodes, OUTC);
}